// MINDLoss_18700287607480
// MI455X (gfx1250) — compile-verified
//
#include <hip/hip_runtime.h>
#include <hip/hip_bf16.h>
#include <math.h>

// MIND loss, fully fused for MI455X (gfx1250, wave32).
// One wave = one 16x16 output tile of the cropped 370x370 descriptor map.
// The 7x7 separable Gaussian conv of each shifted squared-difference map is
// done as two banded matmuls on V_WMMA_F32_16X16X4_F32:
//   T1(22x16) = D2(22x22) x G(22x16)   (row conv; A built directly from LDS tile)
//   Out(16x16) = G'(16x22) x T1(22x16) (col conv)
// All operand loads are unconditional: K-padding is absorbed by zeros in the
// Gaussian operand, M-padding by overlapping M-groups (rows 0..15 and 6..21).

typedef float v2f __attribute__((ext_vector_type(2)));
typedef float v8f __attribute__((ext_vector_type(8)));

#define IMG_H 384
#define OUT_W 370          // 384 - 2*7
#define TS 16
#define NTILE 24           // ceil(370/16)
#define BATCH 4
#define EPSV 1e-5f

// 1D separable Gaussian: exp(-(t-3)^2/(2*2^2)) / sqrt(2*pi*2^2)
__device__ __constant__ float c_g1[7] = {
    0.06475879f, 0.12098537f, 0.17603266f, 0.19947114f,
    0.17603266f, 0.12098537f, 0.06475879f};

// One Gaussian-filtered squared-difference map D_s for shift (sx,sy).
// Result in WMMA C/D layout: VGPR i, lanes 0-15 -> row i, lanes 16-31 ->
// row 8+i; col = lane&15.
__device__ inline v8f compute_D(const float (&in)[32][33],
                                float (&t1t)[16][26],
                                int sx, int sy,
                                const v2f gw[6], int ln)
{
    const int lh = ln >> 4;   // lane half: selects K pair / M half
    const int ll = ln & 15;

    // ---- Step 1 (row conv): two overlapping M-groups, rows 0..15 and 6..21.
    // A entries computed directly: d2[row][k] = (in[row+4][k+4] -
    // in[row+4-sy][k+4-sx])^2.  k up to 23 reads real (finite) data; the
    // matching rows of gw are zero so K-pad contributes nothing.
#pragma unroll
    for (int mg = 0; mg < 2; ++mg) {
        const int row = ll + 6 * mg;                // 0..15 / 6..21 (< 22)
        const float *rp = &in[row + 4][4];
        const float *rs = &in[row + 4 - sy][4 - sx];
        v8f acc = {0.f, 0.f, 0.f, 0.f, 0.f, 0.f, 0.f, 0.f};
#pragma unroll
        for (int kc = 0; kc < 6; ++kc) {
            const int ka = 4 * kc + 2 * lh;
            const float d0 = rp[ka] - rs[ka];
            const float d1 = rp[ka + 1] - rs[ka + 1];
            v2f a;
            a.x = d0 * d0;
            a.y = d1 * d1;
            acc = __builtin_amdgcn_wmma_f32_16x16x4_f32(
                false, a, false, gw[kc], (short)0, acc, false, false);
        }
        // scatter D-layout result into transposed T1: t1t[col][k].
        // Overlapping cols 6..15 get identical values from both groups.
#pragma unroll
        for (int i = 0; i < 8; ++i)
            t1t[ll][6 * mg + 8 * lh + i] = acc[i];
    }
    __syncthreads();

    // ---- Step 2 (col conv): Out = G'(16x24) x T1(24x16); gw zeros kill the
    // k=22..23 pad (t1t cols 22..23 zeroed once at kernel start).
    v8f acc2 = {0.f, 0.f, 0.f, 0.f, 0.f, 0.f, 0.f, 0.f};
#pragma unroll
    for (int kc = 0; kc < 6; ++kc) {
        const int ka = 4 * kc + 2 * lh;
        v2f b;
        b.x = t1t[ll][ka];          // aligned contiguous pair -> ds_load_b64
        b.y = t1t[ll][ka + 1];
        acc2 = __builtin_amdgcn_wmma_f32_16x16x4_f32(
            false, gw[kc], false, b, (short)0, acc2, false, false);
    }
    __syncthreads();   // t1t reads complete before next call's scatter
    return acc2;
}

// Pass over all 99 shifts of one image: per-pixel min_s D and V (unit shifts).
__device__ inline void min_var_pass(const float (&in)[32][33],
                                    float (&t1t)[16][26],
                                    const v2f gw[6], int ln,
                                    v8f &mn_out, v8f &v_out)
{
    v8f mn = {3.4e38f, 3.4e38f, 3.4e38f, 3.4e38f,
              3.4e38f, 3.4e38f, 3.4e38f, 3.4e38f};
    v8f vv = {0.f, 0.f, 0.f, 0.f, 0.f, 0.f, 0.f, 0.f};
#pragma unroll 1
    for (int sy = -5; sy < 5; ++sy) {
#pragma unroll 1
        for (int sx = -5; sx < 5; ++sx) {
            if (sx == 0 && sy == 0) continue;
            v8f D = compute_D(in, t1t, sx, sy, gw, ln);
#pragma unroll
            for (int i = 0; i < 8; ++i) mn[i] = fminf(mn[i], D[i]);
            if (sx * sx + sy * sy == 1) {
#pragma unroll
                for (int i = 0; i < 8; ++i) vv[i] += D[i];
            }
        }
    }
    mn_out = mn;
#pragma unroll
    for (int i = 0; i < 8; ++i) v_out[i] = vv[i] * 0.25f + EPSV;
}

__global__ __launch_bounds__(32)
void mind_loss_tiles(const float *__restrict__ pred,
                     const float *__restrict__ gt,
                     float *__restrict__ partial)
{
    __shared__ float inP[32][33];
    __shared__ float inG[32][33];
    __shared__ float t1t[16][26];

    const int ln = threadIdx.x;
    const int tc = blockIdx.x, tr = blockIdx.y, bz = blockIdx.z;
    const int r0 = tr * TS;    // buffer row 0 == image row r0 (crop 7 - halo 7)
    const int c0 = tc * TS;
    const size_t ib = (size_t)bz * IMG_H * IMG_H;

    // load 32x33 input tiles with wrap (matches jnp.roll); col 32 is reachable
    // as K-pad data (ka=23, sx=-5) so it must be real/finite too.
    for (int idx = ln; idx < 32 * 32; idx += 32) {
        const int a = idx >> 5, b = idx & 31;
        const int gr = (r0 + a) % IMG_H;
        const int gc = (c0 + b) % IMG_H;
        inP[a][b] = pred[ib + (size_t)gr * IMG_H + gc];
        inG[a][b] = gt[ib + (size_t)gr * IMG_H + gc];
    }
    {
        const int gr = (r0 + ln) % IMG_H;
        const int gc = (c0 + 32) % IMG_H;
        inP[ln][32] = pred[ib + (size_t)gr * IMG_H + gc];
        inG[ln][32] = gt[ib + (size_t)gr * IMG_H + gc];
    }
    // zero T1 K-pad columns once (never rewritten; 0 * finite = 0 in step 2)
    if (ln < 16) {
        t1t[ln][22] = 0.f; t1t[ln][23] = 0.f;
        t1t[ln][24] = 0.f; t1t[ln][25] = 0.f;
    }

    // banded Gaussian operand chunks (B in step 1 == A in step 2 per-lane):
    // value = g1[k - (lane&15)] inside the band, zero outside / for k >= 22.
    v2f gw[6];
    {
        const int lh = ln >> 4, ll = ln & 15;
#pragma unroll
        for (int kc = 0; kc < 6; ++kc) {
            const int ka = 4 * kc + 2 * lh;
            const int e0 = ka - ll, e1 = ka + 1 - ll;
            gw[kc].x = (ka < 22 && e0 >= 0 && e0 < 7) ? c_g1[e0] : 0.0f;
            gw[kc].y = (ka + 1 < 22 && e1 >= 0 && e1 < 7) ? c_g1[e1] : 0.0f;
        }
    }
    __syncthreads();

    v8f minP, vP, minG, vG;
    min_var_pass(inP, t1t, gw, ln, minP, vP);
    min_var_pass(inG, t1t, gw, ln, minG, vG);

    // joint pass: M_norm = exp(-(D - minD)/V); accumulate L1 difference.
    v8f loss = {0.f, 0.f, 0.f, 0.f, 0.f, 0.f, 0.f, 0.f};
#pragma unroll 1
    for (int sy = -5; sy < 5; ++sy) {
#pragma unroll 1
        for (int sx = -5; sx < 5; ++sx) {
            if (sx == 0 && sy == 0) continue;
            v8f Dp = compute_D(inP, t1t, sx, sy, gw, ln);
            v8f Dg = compute_D(inG, t1t, sx, sy, gw, ln);
#pragma unroll
            for (int i = 0; i < 8; ++i) {
                const float mp = __expf(-(Dp[i] - minP[i]) / vP[i]);
                const float mg = __expf(-(Dg[i] - minG[i]) / vG[i]);
                loss[i] += fabsf(mp - mg);
            }
        }
    }

    // mask pixels overhanging the 370x370 crop, reduce within the wave
    float s = 0.0f;
#pragma unroll
    for (int i = 0; i < 8; ++i) {
        const int Rr = r0 + 8 * (ln >> 4) + i;  // C/D layout: M = 8*laneHalf+i
        const int Cc = c0 + (ln & 15);
        if (Rr < OUT_W && Cc < OUT_W) s += loss[i];
    }
#pragma unroll
    for (int off = 16; off > 0; off >>= 1)
        s += __shfl_down(s, off, 32);
    if (ln == 0)
        partial[((size_t)bz * NTILE + tr) * NTILE + tc] = s;
}

__global__ __launch_bounds__(256)
void reduce_partials(const float *__restrict__ partial, int n,
                     float *__restrict__ out)
{
    __shared__ float red[256];
    float s = 0.0f;
    for (int i = threadIdx.x; i < n; i += 256) s += partial[i];  // fixed order
    red[threadIdx.x] = s;
    __syncthreads();
    for (int st = 128; st > 0; st >>= 1) {
        if ((int)threadIdx.x < st) red[threadIdx.x] += red[threadIdx.x + st];
        __syncthreads();
    }
    if (threadIdx.x == 0)
        out[0] = red[0] * (1.0f / (99.0f * 4.0f * 370.0f * 370.0f));
}

extern "C" void kernel_launch(void *const *d_in, const int *in_sizes, int n_in,
                              void *d_out, int out_size, void *d_ws, size_t ws_size,
                              hipStream_t stream)
{
    const float *pred = (const float *)d_in[0];
    const float *gt = (const float *)d_in[1];
    float *partial = (float *)d_ws;  // NTILE*NTILE*BATCH floats, overwritten each call

    dim3 grid(NTILE, NTILE, BATCH);
    mind_loss_tiles<<<grid, 32, 0, stream>>>(pred, gt, partial);
    reduce_partials<<<1, 256, 0, stream>>>(partial, NTILE * NTILE * BATCH,
                                           (float *)d_out);
}